// MoELinear_14199161880744
// MI455X (gfx1250) — compile-verified
//
#include <hip/hip_runtime.h>

typedef __attribute__((ext_vector_type(2))) float v2f;
typedef __attribute__((ext_vector_type(8))) float v8f;
typedef int i4 __attribute__((vector_size(16)));   // matches builtin's b128 pointee

constexpr int E_   = 8;
constexpr int IN_  = 2048;
constexpr int OUT_ = 2048;
constexpr int T_   = 4096;

constexpr int BM = 128;
constexpr int BN = 128;
constexpr int BK = 16;
constexpr int STRA = 20;    // A LDS row stride (floats): conflict-free frag reads, 16B-aligned rows
constexpr int STRB = 136;   // B LDS row stride (floats): half-wave bank shift = 16
constexpr int NSTAGE = IN_ / BK;   // 128
constexpr int NBUF = 3;            // async pipeline depth

#if defined(__AMDGCN__) && __has_builtin(__builtin_amdgcn_global_load_async_to_lds_b128)
#define HAVE_ASYNC 1
#else
#define HAVE_ASYNC 0
#endif

#if HAVE_ASYNC
typedef __attribute__((address_space(1))) i4 gi4;   // global int4
typedef __attribute__((address_space(3))) i4 li4;   // LDS int4

__device__ __forceinline__ void async_cp16(const float* g, float* l) {
    // global -> LDS, 16B per lane, tracked by ASYNCcnt, no VGPR data movement
    __builtin_amdgcn_global_load_async_to_lds_b128(
        (gi4*)(unsigned long long)g,
        (li4*)(unsigned int)(unsigned long long)l,
        0, 0);
}
template <int N>
__device__ __forceinline__ void wait_async() {
#if __has_builtin(__builtin_amdgcn_s_wait_asynccnt)
    __builtin_amdgcn_s_wait_asynccnt(N);
#else
    asm volatile("s_wait_asynccnt %0" :: "n"(N) : "memory");
#endif
}
#endif

__global__ __launch_bounds__(256, 4)
void moe_grouped_gemm(const float* __restrict__ x,
                      const float* __restrict__ w,
                      const float* __restrict__ bias,
                      const int* __restrict__ gsz,
                      float* __restrict__ out)
{
    __shared__ float As[NBUF][BM * STRA];   // 3 * 10240 B
    __shared__ float Bs[NBUF][BK * STRB];   // 3 *  8704 B

    const int tid  = threadIdx.x;
    const int lane = tid & 31;
    const int lo16 = lane & 15;
    const int hi   = lane >> 4;          // half-wave select
    const int wave = tid >> 5;           // 0..7
    const int wm   = wave & 1;           // row group of 64
    const int wn   = wave >> 1;          // col group of 32

    const int r0 = blockIdx.y * BM;
    const int n0 = blockIdx.x * BN;

    int startTok = 0;
    for (int e = 0; e < E_; ++e) {
        const int endTok = startTok + gsz[e];
        if (endTok > r0 && startTok < r0 + BM) {
            const float* We = w + (size_t)e * IN_ * OUT_;

            v8f acc[4][2];
            #pragma unroll
            for (int mt = 0; mt < 4; ++mt)
                #pragma unroll
                for (int nt = 0; nt < 2; ++nt)
                    acc[mt][nt] = (v8f){};

            auto load_a = [&](const float* Ac, int kk, v2f (&af)[4]) {
                #pragma unroll
                for (int mt = 0; mt < 4; ++mt)
                    af[mt] = *(const v2f*)(Ac + (wm * 64 + mt * 16 + lo16) * STRA + kk + 2 * hi);
            };
            auto load_b = [&](const float* Bc, int kk, v2f (&bf)[2]) {
                #pragma unroll
                for (int nt = 0; nt < 2; ++nt) {
                    const int nc = wn * 32 + nt * 16 + lo16;
                    bf[nt].x = Bc[(kk     + 2 * hi) * STRB + nc];
                    bf[nt].y = Bc[(kk + 1 + 2 * hi) * STRB + nc];
                }
            };
            auto wmma8 = [&](const v2f (&af)[4], const v2f (&bf)[2]) {
                #pragma unroll
                for (int mt = 0; mt < 4; ++mt)
                    #pragma unroll
                    for (int nt = 0; nt < 2; ++nt)
                        acc[mt][nt] = __builtin_amdgcn_wmma_f32_16x16x4_f32(
                            false, af[mt], false, bf[nt],
                            (short)0, acc[mt][nt], false, false);
            };
            auto compute_stage = [&](const float* Ac, const float* Bc) {
                v2f a0[4], b0[2], a1[4], b1[2];
                load_a(Ac, 0, a0);  load_b(Bc, 0, b0);
                load_a(Ac, 4, a1);  load_b(Bc, 4, b1);
                wmma8(a0, b0);
                load_a(Ac, 8, a0);  load_b(Bc, 8, b0);
                wmma8(a1, b1);
                load_a(Ac, 12, a1); load_b(Bc, 12, b1);
                wmma8(a0, b0);
                wmma8(a1, b1);
            };

            __syncthreads();   // LDS reuse safety across expert passes

#if HAVE_ASYNC
            auto issue_stage = [&](int ks, int buf) {
                const int k0 = ks * BK;
                #pragma unroll
                for (int i = 0; i < 2; ++i) {
                    const int arow = (tid >> 2) + i * 64;      // 128 rows x 4 float4
                    const int ak   = (tid & 3) << 2;
                    async_cp16(x + (size_t)(r0 + arow) * IN_ + k0 + ak,
                               &As[buf][arow * STRA + ak]);
                    const int bkr = (tid >> 5) + i * 8;        // 16 rows x 32 float4
                    const int bn  = (tid & 31) << 2;
                    async_cp16(We + (size_t)(k0 + bkr) * OUT_ + n0 + bn,
                               &Bs[buf][bkr * STRB + bn]);
                }
            };

            // 3-deep pipeline: stages ks (resident), ks+1 (in flight), ks+2 (issued below)
            issue_stage(0, 0);
            issue_stage(1, 1);
            for (int ks = 0; ks < NSTAGE; ++ks) {
                // 4 async ops per wave per stage, in-order completion:
                // leave the newest stage outstanding, require the oldest to be done.
                if (ks + 1 < NSTAGE) wait_async<4>();
                else                 wait_async<0>();
                __syncthreads();
                if (ks + 2 < NSTAGE)
                    issue_stage(ks + 2, (ks + 2) % NBUF);
                compute_stage(As[ks % NBUF], Bs[ks % NBUF]);
            }
#else
            // synchronous fallback: single-buffer staging
            for (int ks = 0; ks < NSTAGE; ++ks) {
                const int k0 = ks * BK;
                float4 ra[2], rb[2];
                #pragma unroll
                for (int i = 0; i < 2; ++i) {
                    const int arow = (tid >> 2) + i * 64;
                    const int ak   = (tid & 3) << 2;
                    const int bkr  = (tid >> 5) + i * 8;
                    const int bn   = (tid & 31) << 2;
                    ra[i] = *(const float4*)(x  + (size_t)(r0 + arow) * IN_ + k0 + ak);
                    rb[i] = *(const float4*)(We + (size_t)(k0 + bkr) * OUT_ + n0 + bn);
                }
                __syncthreads();
                #pragma unroll
                for (int i = 0; i < 2; ++i) {
                    const int arow = (tid >> 2) + i * 64;
                    const int ak   = (tid & 3) << 2;
                    const int bkr  = (tid >> 5) + i * 8;
                    const int bn   = (tid & 31) << 2;
                    *(float4*)&As[0][arow * STRA + ak] = ra[i];
                    *(float4*)&Bs[0][bkr * STRB + bn] = rb[i];
                }
                __syncthreads();
                compute_stage(As[0], Bs[0]);
            }
#endif

            // ---- epilogue: write only rows owned by expert e ----
            const int lo    = startTok > r0 ? startTok : r0;
            const int hiRow = endTok < r0 + BM ? endTok : r0 + BM;
            #pragma unroll
            for (int nt = 0; nt < 2; ++nt) {
                const int col = n0 + wn * 32 + nt * 16 + lo16;
                const float bv = bias[col];
                #pragma unroll
                for (int mt = 0; mt < 4; ++mt) {
                    const int rowBase = r0 + wm * 64 + mt * 16 + hi * 8;
                    #pragma unroll
                    for (int v = 0; v < 8; ++v) {
                        const int row = rowBase + v;   // C/D layout: elem v -> M = v + 8*hi
                        if (row >= lo && row < hiRow)
                            out[(size_t)row * OUT_ + col] = acc[mt][nt][v] + bv;
                    }
                }
            }
        }
        startTok = endTok;
    }
}

extern "C" void kernel_launch(void* const* d_in, const int* in_sizes, int n_in,
                              void* d_out, int out_size, void* d_ws, size_t ws_size,
                              hipStream_t stream) {
    const float* x    = (const float*)d_in[0];
    const float* w    = (const float*)d_in[1];
    const float* bias = (const float*)d_in[2];
    const int*   gsz  = (const int*)d_in[3];
    float* out = (float*)d_out;

    dim3 grid(OUT_ / BN, T_ / BM);   // (16, 32)
    moe_grouped_gemm<<<grid, dim3(256), 0, stream>>>(x, w, bias, gsz, out);
}